// ESN_52441550684603
// MI455X (gfx1250) — compile-verified
//
#include <hip/hip_runtime.h>
#include <math.h>

#define ESN_N       4096
#define ESN_T       4096
#define ESN_WASHOUT 200
#define NBLK        128   // persistent blocks; 128 x 256 threads is resident on any MI-class part
#define NTH         256   // 8 waves of 32

typedef float v2f __attribute__((ext_vector_type(2)));
typedef float v8f __attribute__((ext_vector_type(8)));

// ---------------------------------------------------------------------------
// Init: zero x double-buffer, barrier counter, and d_out (harness poisons once
// with 0xAA and never re-poisons, so every launch must self-initialize).
// ---------------------------------------------------------------------------
__global__ void esn_init_kernel(float* __restrict__ xbuf,
                                float* __restrict__ out,
                                unsigned* __restrict__ bar) {
    int i = blockIdx.x * blockDim.x + threadIdx.x;
    int stride = gridDim.x * blockDim.x;
    for (int j = i; j < 2 * ESN_N; j += stride) xbuf[j] = 0.0f;
    for (int j = i; j < ESN_T - ESN_WASHOUT; j += stride) out[j] = 0.0f;
    if (i == 0) *bar = 0u;
}

// ---------------------------------------------------------------------------
// Persistent ESN rollout. W stays L2-resident (64 MB < 192 MB L2).
// Per step: GEMV via V_WMMA_F32_16X16X4_F32 (fp32-exact), tanh, grid barrier,
// deterministic readout by block 0.
//
// Layout per block (32 rows of W):
//   wave = tid>>5 (0..7); group = wave>>2 (16-row group); kw = wave&3 (K chunk)
//   Each wave: A tile = 16 rows x 4 cols of W, K swept over its 1024-wide chunk.
//   A VGPR layout (ISA 7.12.2): lanes 0-15 hold K={0,1}, lanes 16-31 K={2,3}
//   -> lane loads float2 from W[row][k0 + 2*(lane>=16)]  (contiguous row stream)
//   B = x[k0..k0+3] broadcast across all 16 columns -> every D column equals
//   the 16 row-partials; lane 0 carries rows 0-7 (v0..v7), lane 16 rows 8-15.
// ---------------------------------------------------------------------------
__global__ __launch_bounds__(NTH, 1) void esn_persistent_kernel(
    const float* __restrict__ u,      // [T]
    const float* __restrict__ W,      // [N][N] row-major
    const float* __restrict__ w_in,   // [N]
    const float* __restrict__ w_out,  // [N]
    const int*   __restrict__ mask,   // [N]
    float*       __restrict__ out,    // [T - WASHOUT]
    float*       __restrict__ xbuf,   // [2][N] double buffer (workspace)
    unsigned*    __restrict__ bar)    // [1]   barrier counter (workspace)
{
    const int tid  = threadIdx.x;
    const int wave = tid >> 5;        // 0..7
    const int lane = tid & 31;
    const int grp  = wave >> 2;       // 0..1  : which 16-row group
    const int kw   = wave & 3;        // 0..3  : which 1024-wide K chunk
    const int m    = lane & 15;       // row within group
    const int hi   = lane >> 4;       // 0/1   : K sub-offset {0,1} vs {2,3}

    const int    row     = blockIdx.x * 32 + grp * 16 + m;
    const int    kChunk  = kw * (ESN_N / 4) + hi * 2;
    const float* wrowBase = W + (size_t)row * ESN_N + kChunk;

    __shared__ float red[8][16];      // per-wave 16-row partials
    __shared__ float rsum[NTH];       // block-0 readout reduction

    for (int t = 0; t < ESN_T; ++t) {
        const float* xin  = xbuf + ((t)     & 1) * ESN_N;
        float*       xout = xbuf + ((t + 1) & 1) * ESN_N;
        const float* xk   = xin + kChunk;

        // ---- GEMV partial: 256 x (b64 W-load, b64 x-load, f32 WMMA) ----
        v8f d = {};
        #pragma unroll 4
        for (int kk = 0; kk < ESN_N / 4; kk += 4) {
            v2f a = *(const v2f*)(wrowBase + kk);   // W[row][k0+2*hi .. +1]
            v2f b = *(const v2f*)(xk + kk);         // x[k0+2*hi .. +1] (bcast over cols)
            d = __builtin_amdgcn_wmma_f32_16x16x4_f32(
                    false, a, false, b, (short)0, d, false, false);
        }

        // D columns are identical: lane 0 holds rows 0-7, lane 16 rows 8-15.
        if (m == 0) {
            #pragma unroll
            for (int r = 0; r < 8; ++r) red[wave][hi * 8 + r] = d[r];
        }
        __syncthreads();

        // ---- combine 4 K-chunk waves, apply tanh, write x_{t+1} ----
        if (tid < 32) {
            const int g4 = (tid >> 4) << 2;   // base wave of this group
            const int r  = tid & 15;
            float s = red[g4][r] + red[g4 + 1][r] + red[g4 + 2][r] + red[g4 + 3][r];
            const int ri = blockIdx.x * 32 + tid;
            xout[ri] = tanhf(w_in[ri] * u[t] + s);
        }

        // ---- device-wide barrier (release writes, acquire before reads) ----
        __threadfence();
        __syncthreads();
        if (tid == 0) {
            __hip_atomic_fetch_add(bar, 1u, __ATOMIC_RELEASE, __HIP_MEMORY_SCOPE_AGENT);
            const unsigned tgt = (unsigned)(t + 1) * (unsigned)NBLK;
            while (__hip_atomic_load(bar, __ATOMIC_ACQUIRE, __HIP_MEMORY_SCOPE_AGENT) < tgt) {
                __builtin_amdgcn_s_sleep(2);
            }
        }
        __syncthreads();

        // ---- readout: out[t-200] = sum_n w_out[n] * x_{t+1}[mask[n]] ----
        // Single block + fixed-order tree => bit-deterministic across replays.
        if (t >= ESN_WASHOUT && blockIdx.x == 0) {
            float p = 0.0f;
            for (int n = tid; n < ESN_N; n += NTH)
                p += w_out[n] * xout[mask[n]];
            rsum[tid] = p;
            __syncthreads();
            #pragma unroll
            for (int off = NTH / 2; off > 0; off >>= 1) {
                if (tid < off) rsum[tid] += rsum[tid + off];
                __syncthreads();
            }
            if (tid == 0) out[t - ESN_WASHOUT] = rsum[0];
        }
    }
}

// ---------------------------------------------------------------------------
extern "C" void kernel_launch(void* const* d_in, const int* in_sizes, int n_in,
                              void* d_out, int out_size, void* d_ws, size_t ws_size,
                              hipStream_t stream) {
    const float* u     = (const float*)d_in[0];   // [4096]
    const float* w_res = (const float*)d_in[1];   // [4096*4096]
    const float* w_in  = (const float*)d_in[2];   // [4096]
    const float* w_out = (const float*)d_in[3];   // [4096]
    const int*   mask  = (const int*)d_in[4];     // [4096]
    float*       out   = (float*)d_out;           // [3896]

    float*    xbuf = (float*)d_ws;                           // 2*N floats
    unsigned* bar  = (unsigned*)((char*)d_ws + 2 * ESN_N * sizeof(float));

    esn_init_kernel<<<32, 256, 0, stream>>>(xbuf, out, bar);
    esn_persistent_kernel<<<NBLK, NTH, 0, stream>>>(u, w_res, w_in, w_out, mask,
                                                    out, xbuf, bar);
}